// ExponentialRepulsionBlock_41798621724834
// MI455X (gfx1250) — compile-verified
//
#include <hip/hip_runtime.h>

#define ALPHA 3.0f
// exp(-a*x) == exp2(-a*log2(e)*x) -> single v_exp_f32
#define NEG_ALPHA_LOG2E (-ALPHA * 1.44269504088896340736f)

typedef int   __attribute__((ext_vector_type(4))) i32x4;
typedef float __attribute__((ext_vector_type(4))) f32x4;

__global__ void __launch_bounds__(256)
zero_f32_kernel(float* __restrict__ out, int n) {
  int i = blockIdx.x * 256 + threadIdx.x;
  if (i < n) out[i] = 0.0f;
}

__device__ __forceinline__ void edge_one(const float* __restrict__ pos,
                                         float* __restrict__ out,
                                         int s, int r,
                                         float shx, float shy, float shz) {
  // Gathers: 1.2 MB positions table is L2-resident (192 MB L2); each gather
  // merges to a single global_load_b96.
  const float dx = pos[3 * r + 0] - pos[3 * s + 0] + shx;
  const float dy = pos[3 * r + 1] - pos[3 * s + 1] + shy;
  const float dz = pos[3 * r + 2] - pos[3 * s + 2] + shz;

  const float d2  = __builtin_fmaf(dx, dx, __builtin_fmaf(dy, dy, dz * dz));
  const float len = __builtin_amdgcn_sqrtf(d2);                            // v_sqrt_f32
  const float en  = 0.5f * __builtin_amdgcn_exp2f(NEG_ALPHA_LOG2E * len);  // v_exp_f32

  // Native FP32 atomic (global_atomic_add_f32, no-return -> STOREcnt),
  // ~64 updates/node avg resolved in the L2 atomic units.
  unsafeAtomicAdd(&out[s], en);
}

__global__ void __launch_bounds__(256)
edge_energy_kernel(const float* __restrict__ pos,
                   const int*   __restrict__ sender,
                   const int*   __restrict__ receiver,
                   const float* __restrict__ shifts,
                   float*       __restrict__ out,
                   int num_edges) {
  const int t  = blockIdx.x * 256 + threadIdx.x;
  const int e0 = t * 4;
  if (e0 >= num_edges) return;

  // gfx1250 global_prefetch_b8: fire-and-forget (no LOADcnt), ~16K edges ahead
  // on each streamed array to deepen the HBM prefetch window.
  const int pe = e0 + 16384;
  if (pe < num_edges) {
    __builtin_prefetch(&sender[pe],     0, 1);
    __builtin_prefetch(&receiver[pe],   0, 1);
    __builtin_prefetch(&shifts[3 * pe], 0, 1);
  }

  if (e0 + 3 < num_edges) {
    // Streamed traffic in b128 quanta: 2x idx + 3x shifts per 4 edges.
    const i32x4 s4 = *(const i32x4*)(sender + e0);    // global_load_b128
    const i32x4 r4 = *(const i32x4*)(receiver + e0);  // global_load_b128
    const f32x4* sv = (const f32x4*)(shifts + 3 * e0);
    const f32x4 a = sv[0];                            // global_load_b128 x3
    const f32x4 b = sv[1];
    const f32x4 c = sv[2];

    const float sh[12] = {a.x, a.y, a.z, a.w,
                          b.x, b.y, b.z, b.w,
                          c.x, c.y, c.z, c.w};
#pragma unroll
    for (int j = 0; j < 4; ++j) {
      edge_one(pos, out, s4[j], r4[j], sh[3 * j + 0], sh[3 * j + 1], sh[3 * j + 2]);
    }
  } else {
    // Scalar tail (E % 4 != 0 case; unused for E = 6.4M but keeps it general).
    for (int e = e0; e < num_edges; ++e) {
      edge_one(pos, out, sender[e], receiver[e],
               shifts[3 * e + 0], shifts[3 * e + 1], shifts[3 * e + 2]);
    }
  }
}

extern "C" void kernel_launch(void* const* d_in, const int* in_sizes, int n_in,
                              void* d_out, int out_size, void* d_ws, size_t ws_size,
                              hipStream_t stream) {
  const float* pos    = (const float*)d_in[0];
  const int*   ei     = (const int*)d_in[1];   // [2, E] flat: senders, then receivers
  const float* shifts = (const float*)d_in[2];
  float*       out    = (float*)d_out;

  const int E = in_sizes[2] / 3;   // 6,400,000 (derived from shifts, index-width agnostic)
  const int N = out_size;          // 100,000

  const int* sender   = ei;
  const int* receiver = ei + E;

  // Output is accumulated into and the harness poisons it once: zero it every call.
  zero_f32_kernel<<<(N + 255) / 256, 256, 0, stream>>>(out, N);

  const int threads = (E + 3) / 4;
  edge_energy_kernel<<<(threads + 255) / 256, 256, 0, stream>>>(
      pos, sender, receiver, shifts, out, E);
}